// GRU_GAT_27693949124839
// MI455X (gfx1250) — compile-verified
//
#include <hip/hip_runtime.h>
#include <math.h>

typedef __attribute__((ext_vector_type(2))) float v2f;
typedef __attribute__((ext_vector_type(4))) float v4f;
typedef __attribute__((ext_vector_type(8))) float v8f;

#define GRU_T 16
#define LDS_STRIDE 66   // 16x64 tile padded to avoid 64-bank conflicts (stride 66 dwords, even for b64)

// ---------------------------------------------------------------------------
// WMMA f32 16x16x4 wrapper (fp32 A/B/C — full precision for the GRU recurrence)
// ---------------------------------------------------------------------------
__device__ __forceinline__ v8f wmma4(v2f a, v2f b, v8f c) {
  return __builtin_amdgcn_wmma_f32_16x16x4_f32(
      /*neg_a=*/false, a, /*neg_b=*/false, b,
      /*c_mod=*/(short)0, c, /*reuse_a=*/false, /*reuse_b=*/false);
}

// A fragment (16x4) from an LDS 16xSTRIDE row-major tile.
// lanes 0-15: M=lane, K = 4kc+0,1 ; lanes 16-31: M=lane-16, K = 4kc+2,3
__device__ __forceinline__ v2f lds_afrag(const float* buf, int lane, int kc) {
  int half = lane >> 4;
  int m    = lane & 15;
  const float* p = buf + m * LDS_STRIDE + kc * 4 + half * 2;
  v2f r; r[0] = p[0]; r[1] = p[1];
  return r;
}

// B fragment (4x16) from the pre-packed GRU weight buffer (kc-major layout).
// Packed layout: Wpk[((kc*192 + n)*2 + half)*2 + q] = W[n][4kc + 2*half + q]
__device__ __forceinline__ v2f bfrag_packed(const float* Wpk, int lane, int kc, int j) {
  int half = lane >> 4;
  int n    = j * 16 + (lane & 15);
  const float* p = Wpk + (((size_t)(kc * 192 + n) * 2 + half) << 1);
  v2f r; r[0] = p[0]; r[1] = p[1];
  return r;
}

// Scatter C-layout regs (4 tiles of 16x16) into an LDS 16xSTRIDE row-major tile.
__device__ __forceinline__ void store_c_to_lds(float* buf, const v8f* hC, int lane) {
  int mh = 8 * (lane >> 4);
  int lc = lane & 15;
#pragma unroll
  for (int j = 0; j < 4; j++) {
#pragma unroll
    for (int r = 0; r < 8; r++) {
      buf[(mh + r) * LDS_STRIDE + j * 16 + lc] = hC[j][r];
    }
  }
}

// ---------------------------------------------------------------------------
// One GRU step for one 16-node tile (M=16, K=64, gates=192 -> 12 n-tiles).
// r/z gates accumulate gi+gh jointly (8 accs); n gate kept split (4+4 accs).
// kc is a RUNTIME loop (no hoisting across t); the 24 weight fragments are
// batched into register arrays so all loads are in flight before the WMMA
// burst (one wait per 24 WMMAs instead of one wait per WMMA).
// ---------------------------------------------------------------------------
__device__ __forceinline__ void gru_step(
    const float* aIn, const float* hBuf,
    const float* __restrict__ Wpki, const float* __restrict__ Wpkh,
    const float* brz, const float* bin_, const float* bhn,
    v8f* hC, int lane)
{
  v8f rz[8], gin[4], ghn[4];
#pragma unroll
  for (int j = 0; j < 8; j++) {
#pragma unroll
    for (int r = 0; r < 8; r++) rz[j][r] = brz[j];
  }
#pragma unroll
  for (int j = 0; j < 4; j++) {
#pragma unroll
    for (int r = 0; r < 8; r++) { gin[j][r] = bin_[j]; ghn[j][r] = bhn[j]; }
  }

#pragma unroll 1
  for (int kc = 0; kc < 16; kc++) {
    v2f ax = lds_afrag(aIn,  lane, kc);
    v2f ah = lds_afrag(hBuf, lane, kc);
    v2f bi[12], bh[12];
#pragma unroll
    for (int j = 0; j < 12; j++) {
      bi[j] = bfrag_packed(Wpki, lane, kc, j);
      bh[j] = bfrag_packed(Wpkh, lane, kc, j);
    }
#pragma unroll
    for (int j = 0; j < 8; j++) {
      rz[j] = wmma4(ax, bi[j], rz[j]);
      rz[j] = wmma4(ah, bh[j], rz[j]);
    }
#pragma unroll
    for (int j = 0; j < 4; j++) {
      gin[j] = wmma4(ax, bi[8 + j], gin[j]);
      ghn[j] = wmma4(ah, bh[8 + j], ghn[j]);
    }
  }

#pragma unroll
  for (int j = 0; j < 4; j++) {
#pragma unroll
    for (int r = 0; r < 8; r++) {
      float rg = 1.0f / (1.0f + __expf(-rz[j][r]));
      float zg = 1.0f / (1.0f + __expf(-rz[j + 4][r]));
      float ng = tanhf(gin[j][r] + rg * ghn[j][r]);
      hC[j][r] = (1.0f - zg) * ng + zg * hC[j][r];
    }
  }
}

// ---------------------------------------------------------------------------
// Fused 2-layer GRU over all T steps; one wave owns a 16-node tile.
// Layer-2 consumes layer-1's step output straight from LDS — no HBM traffic
// for intermediates. Emits feats = h2[:, T-1, :].
// ---------------------------------------------------------------------------
__global__ __launch_bounds__(128) void gru_fused(
    const float* __restrict__ x,
    const float* __restrict__ wpk0i, const float* __restrict__ wpk0h,
    const float* __restrict__ wpk1i, const float* __restrict__ wpk1h,
    const float* __restrict__ bih0, const float* __restrict__ bhh0,
    const float* __restrict__ bih1, const float* __restrict__ bhh1,
    float* __restrict__ feats, int Nn)
{
  __shared__ float lds[4][2][16 * LDS_STRIDE];
  int wave = threadIdx.x >> 5;
  int lane = threadIdx.x & 31;
  int tile = blockIdx.x * 4 + wave;
  if (tile * 16 >= Nn) return;
  int m0 = tile * 16;
  float* aIn  = &lds[wave][0][0];
  float* hBuf = &lds[wave][1][0];

  int lc = lane & 15;
  float brz0[8], brz1[8], bin0[4], bin1[4], bhn0[4], bhn1[4];
#pragma unroll
  for (int j = 0; j < 8; j++) {
    brz0[j] = bih0[j * 16 + lc] + bhh0[j * 16 + lc];
    brz1[j] = bih1[j * 16 + lc] + bhh1[j * 16 + lc];
  }
#pragma unroll
  for (int j = 0; j < 4; j++) {
    bin0[j] = bih0[128 + j * 16 + lc]; bhn0[j] = bhh0[128 + j * 16 + lc];
    bin1[j] = bih1[128 + j * 16 + lc]; bhn1[j] = bhh1[128 + j * 16 + lc];
  }

  v8f h0C[4], h1C[4];
#pragma unroll
  for (int j = 0; j < 4; j++) {
#pragma unroll
    for (int r = 0; r < 8; r++) { h0C[j][r] = 0.0f; h1C[j][r] = 0.0f; }
  }

#pragma unroll 1
  for (int t = 0; t < GRU_T; t++) {
    // stage x[:,t,:] tile into LDS (coalesced 16B global loads)
#pragma unroll
    for (int i = 0; i < 8; i++) {
      int c = lane + 32 * i;
      int m = c >> 4;
      int k = (c & 15) * 4;
      const v4f gv = *(const v4f*)(x + (((size_t)(m0 + m)) * GRU_T + t) * 64 + k);
      v2f lo; lo[0] = gv[0]; lo[1] = gv[1];
      v2f hi; hi[0] = gv[2]; hi[1] = gv[3];
      *(v2f*)&aIn[m * LDS_STRIDE + k]     = lo;
      *(v2f*)&aIn[m * LDS_STRIDE + k + 2] = hi;
    }
    store_c_to_lds(hBuf, h0C, lane);
    gru_step(aIn, hBuf, wpk0i, wpk0h, brz0, bin0, bhn0, h0C, lane);
    store_c_to_lds(aIn,  h0C, lane);   // layer-2 input
    store_c_to_lds(hBuf, h1C, lane);
    gru_step(aIn, hBuf, wpk1i, wpk1h, brz1, bin1, bhn1, h1C, lane);
  }

  int mh = 8 * (lane >> 4);
#pragma unroll
  for (int j = 0; j < 4; j++) {
#pragma unroll
    for (int r = 0; r < 8; r++) {
      feats[(size_t)(m0 + mh + r) * 64 + j * 16 + lc] = h1C[j][r];
    }
  }
}

// Repack a (192,64) row-major weight into the WMMA-B-fragment-native layout.
__global__ void repack_w(const float* __restrict__ W, float* __restrict__ Wpk) {
  int p = blockIdx.x * blockDim.x + threadIdx.x;
  if (p >= 192 * 64) return;
  int q    = p & 1;
  int half = (p >> 1) & 1;
  int n    = (p >> 2) % 192;
  int kc   = (p >> 2) / 192;
  Wpk[p] = W[n * 64 + kc * 4 + half * 2 + q];
}

// ---------------------------------------------------------------------------
// Generic WMMA GEMM: out(MxNOUT) = A(MxK) @ W(NOUTxK)^T.
// All NOUT/16 output tiles accumulate simultaneously; kc is a runtime loop.
// Per iteration: 1 A fragment + NOUT/16 B fragments batched into arrays,
// then a WMMA burst — loads stay in flight across the whole burst.
// ---------------------------------------------------------------------------
template <int K, int NOUT>
__global__ __launch_bounds__(128) void gemm_wt(
    const float* __restrict__ A, const float* __restrict__ W,
    float* __restrict__ out, int M)
{
  constexpr int NT = NOUT / 16;
  int wave = threadIdx.x >> 5;
  int lane = threadIdx.x & 31;
  int tile = blockIdx.x * 4 + wave;
  int m0 = tile * 16;
  if (m0 >= M) return;
  int half = lane >> 4;
  int mrow = lane & 15;
  int lc   = lane & 15;
  int mh   = 8 * half;

  v8f acc[NT];
#pragma unroll
  for (int j = 0; j < NT; j++) {
#pragma unroll
    for (int r = 0; r < 8; r++) acc[j][r] = 0.0f;
  }

#pragma unroll 1
  for (int kc = 0; kc < K / 4; kc++) {
    const float* ap = A + (size_t)(m0 + mrow) * K + kc * 4 + half * 2;
    v2f a; a[0] = ap[0]; a[1] = ap[1];
    v2f b[NT];
#pragma unroll
    for (int j = 0; j < NT; j++) {
      const float* p = W + (size_t)(j * 16 + lc) * K + kc * 4 + half * 2;
      b[j][0] = p[0]; b[j][1] = p[1];
    }
#pragma unroll
    for (int j = 0; j < NT; j++) acc[j] = wmma4(a, b[j], acc[j]);
  }

#pragma unroll
  for (int j = 0; j < NT; j++) {
#pragma unroll
    for (int r = 0; r < 8; r++) {
      out[(size_t)(m0 + mh + r) * NOUT + j * 16 + lc] = acc[j][r];
    }
  }
}

// ---------------------------------------------------------------------------
// GAT helpers
// ---------------------------------------------------------------------------
__global__ void attn_scores_k(const float* __restrict__ hfeat,
                              const float* __restrict__ att_src,
                              const float* __restrict__ att_dst,
                              float* __restrict__ a_src, float* __restrict__ a_dst,
                              int n_nodes, int heads, int ch)
{
  int i = blockIdx.x * blockDim.x + threadIdx.x;
  if (i >= n_nodes * heads) return;
  int nidx = i / heads, h = i % heads;
  const float* hp = hfeat + ((size_t)nidx * heads + h) * ch;
  float s = 0.0f, d = 0.0f;
  for (int c = 0; c < ch; c++) {
    float v = hp[c];
    s += v * att_src[h * ch + c];
    d += v * att_dst[h * ch + c];
  }
  a_src[i] = s;
  a_dst[i] = d;
}

__global__ void fill_k(float* __restrict__ p, float v, size_t n) {
  size_t i = (size_t)blockIdx.x * blockDim.x + threadIdx.x;
  if (i < n) p[i] = v;
}

__global__ void out_bias_k(float* __restrict__ out, const float* __restrict__ bias,
                           size_t tot, int ch) {
  size_t i = (size_t)blockIdx.x * blockDim.x + threadIdx.x;
  if (i < tot) out[i] = bias[i % ch];
}

__global__ void bias_relu_k(float* __restrict__ g, const float* __restrict__ bias,
                            size_t tot, int ch) {
  size_t i = (size_t)blockIdx.x * blockDim.x + threadIdx.x;
  if (i < tot) g[i] = fmaxf(g[i] + bias[i % ch], 0.0f);
}

__device__ __forceinline__ void get_edge(const int* src, const int* dst,
                                         long e, long Ne, int& s, int& d) {
  if (e < Ne) { s = src[e]; d = dst[e]; }
  else        { s = d = (int)(e - Ne); }   // implicit self-loop
}

__device__ __forceinline__ void atomicMaxF(float* addr, float val) {
  int* ai = (int*)addr;
  int old = __float_as_int(*addr);
  while (__int_as_float(old) < val) {
    int assumed = old;
    old = atomicCAS(ai, assumed, __float_as_int(val));
    if (old == assumed) break;
  }
}

__global__ void edge_max_k(const int* __restrict__ src, const int* __restrict__ dst,
                           int Ne, int Nn, int heads,
                           const float* __restrict__ a_src, const float* __restrict__ a_dst,
                           float* __restrict__ amax)
{
  size_t i = (size_t)blockIdx.x * blockDim.x + threadIdx.x;
  size_t tot = (size_t)(Ne + Nn) * heads;
  if (i >= tot) return;
  long e = (long)(i / heads);
  int  h = (int)(i % heads);
  int s, d;
  get_edge(src, dst, e, Ne, s, d);
  float al = a_src[(size_t)s * heads + h] + a_dst[(size_t)d * heads + h];
  al = al > 0.0f ? al : 0.2f * al;          // leaky_relu, NEG_SLOPE=0.2
  atomicMaxF(&amax[(size_t)d * heads + h], al);
}

__global__ void edge_expsum_k(const int* __restrict__ src, const int* __restrict__ dst,
                              int Ne, int Nn, int heads,
                              const float* __restrict__ a_src, const float* __restrict__ a_dst,
                              const float* __restrict__ amax,
                              float* __restrict__ ebuf, float* __restrict__ denom)
{
  size_t i = (size_t)blockIdx.x * blockDim.x + threadIdx.x;
  size_t tot = (size_t)(Ne + Nn) * heads;
  if (i >= tot) return;
  long e = (long)(i / heads);
  int  h = (int)(i % heads);
  int s, d;
  get_edge(src, dst, e, Ne, s, d);
  float al = a_src[(size_t)s * heads + h] + a_dst[(size_t)d * heads + h];
  al = al > 0.0f ? al : 0.2f * al;
  float e2 = __expf(al - amax[(size_t)d * heads + h]);
  ebuf[i] = e2;
  atomicAdd(&denom[(size_t)d * heads + h], e2);
}

__global__ void edge_msg_k(const int* __restrict__ src, const int* __restrict__ dst,
                           int Ne, int Nn, int heads, int ch,
                           const float* __restrict__ ebuf, const float* __restrict__ denom,
                           const float* __restrict__ hfeat, float* __restrict__ out)
{
  size_t i = (size_t)blockIdx.x * blockDim.x + threadIdx.x;
  int hc = heads * ch;
  size_t tot = (size_t)(Ne + Nn) * hc;
  if (i >= tot) return;
  long e = (long)(i / hc);
  int  c = (int)(i % hc);
  int  h = c / ch;
  int s, d;
  get_edge(src, dst, e, Ne, s, d);
  float e2   = ebuf[(size_t)e * heads + h];
  float coef = e2 / (denom[(size_t)d * heads + h] + 1e-16f);
  atomicAdd(&out[(size_t)d * hc + c], hfeat[(size_t)s * hc + c] * coef);
}

// ---------------------------------------------------------------------------
// Host-side orchestration
// ---------------------------------------------------------------------------
static inline unsigned cdiv(size_t n, unsigned b) { return (unsigned)((n + b - 1) / b); }

extern "C" void kernel_launch(void* const* d_in, const int* in_sizes, int n_in,
                              void* d_out, int out_size, void* d_ws, size_t ws_size,
                              hipStream_t stream)
{
  const float* x     = (const float*)d_in[0];
  const int*   ei    = (const int*)d_in[1];
  const float* Wih0  = (const float*)d_in[2];
  const float* Whh0  = (const float*)d_in[3];
  const float* bih0  = (const float*)d_in[4];
  const float* bhh0  = (const float*)d_in[5];
  const float* Wih1  = (const float*)d_in[6];
  const float* Whh1  = (const float*)d_in[7];
  const float* bih1  = (const float*)d_in[8];
  const float* bhh1  = (const float*)d_in[9];
  const float* W1    = (const float*)d_in[10];
  const float* as1   = (const float*)d_in[11];
  const float* ad1   = (const float*)d_in[12];
  const float* bias1 = (const float*)d_in[13];
  const float* W2    = (const float*)d_in[14];
  const float* as2   = (const float*)d_in[15];
  const float* ad2   = (const float*)d_in[16];
  const float* bias2 = (const float*)d_in[17];

  const int N = in_sizes[0] / (GRU_T * 64);   // 20000
  const int E = in_sizes[1] / 2;              // 640000
  const int* src = ei;
  const int* dst = ei + E;

  float* ws = (float*)d_ws;
  size_t off = 0;
  float* wpk0i = ws + off; off += 192 * 64;
  float* wpk0h = ws + off; off += 192 * 64;
  float* wpk1i = ws + off; off += 192 * 64;
  float* wpk1h = ws + off; off += 192 * 64;
  float* feats = ws + off; off += (size_t)N * 64;
  float* h1    = ws + off; off += (size_t)N * 256;
  float* asr1  = ws + off; off += (size_t)N * 4;
  float* adt1  = ws + off; off += (size_t)N * 4;
  float* amax1 = ws + off; off += (size_t)N * 4;
  float* den1  = ws + off; off += (size_t)N * 4;
  float* ebuf  = ws + off; off += (size_t)(E + N) * 4;
  float* g1    = ws + off; off += (size_t)N * 256;
  float* h2    = ws + off; off += (size_t)N * 64;
  float* asr2  = ws + off; off += (size_t)N;
  float* adt2  = ws + off; off += (size_t)N;
  float* amax2 = ws + off; off += (size_t)N;
  float* den2  = ws + off; off += (size_t)N;
  float* out   = (float*)d_out;

  // 1. repack GRU weights into WMMA-B-fragment layout
  repack_w<<<cdiv(192 * 64, 256), 256, 0, stream>>>(Wih0, wpk0i);
  repack_w<<<cdiv(192 * 64, 256), 256, 0, stream>>>(Whh0, wpk0h);
  repack_w<<<cdiv(192 * 64, 256), 256, 0, stream>>>(Wih1, wpk1i);
  repack_w<<<cdiv(192 * 64, 256), 256, 0, stream>>>(Whh1, wpk1h);

  // 2. fused 2-layer GRU -> feats (N x 64)
  {
    unsigned tiles = (unsigned)((N + 15) / 16);
    gru_fused<<<cdiv(tiles, 4), 128, 0, stream>>>(
        x, wpk0i, wpk0h, wpk1i, wpk1h, bih0, bhh0, bih1, bhh1, feats, N);
  }

  // 3. GAT layer 1 (heads=4, ch=64)
  {
    unsigned tiles = (unsigned)((N + 15) / 16);
    gemm_wt<64, 256><<<cdiv(tiles, 4), 128, 0, stream>>>(feats, W1, h1, N);
    attn_scores_k<<<cdiv((size_t)N * 4, 256), 256, 0, stream>>>(h1, as1, ad1, asr1, adt1, N, 4, 64);
    fill_k<<<cdiv((size_t)N * 4, 256), 256, 0, stream>>>(amax1, -INFINITY, (size_t)N * 4);
    fill_k<<<cdiv((size_t)N * 4, 256), 256, 0, stream>>>(den1, 0.0f, (size_t)N * 4);
    fill_k<<<cdiv((size_t)N * 256, 256), 256, 0, stream>>>(g1, 0.0f, (size_t)N * 256);
    size_t eh = (size_t)(E + N) * 4;
    edge_max_k<<<cdiv(eh, 256), 256, 0, stream>>>(src, dst, E, N, 4, asr1, adt1, amax1);
    edge_expsum_k<<<cdiv(eh, 256), 256, 0, stream>>>(src, dst, E, N, 4, asr1, adt1, amax1, ebuf, den1);
    size_t ec = (size_t)(E + N) * 256;
    edge_msg_k<<<cdiv(ec, 256), 256, 0, stream>>>(src, dst, E, N, 4, 64, ebuf, den1, h1, g1);
    bias_relu_k<<<cdiv((size_t)N * 256, 256), 256, 0, stream>>>(g1, bias1, (size_t)N * 256, 256);
  }

  // 4. GAT layer 2 (heads=1, ch=64) -> d_out
  {
    unsigned tiles = (unsigned)((N + 15) / 16);
    gemm_wt<256, 64><<<cdiv(tiles, 4), 128, 0, stream>>>(g1, W2, h2, N);
    attn_scores_k<<<cdiv((size_t)N, 256), 256, 0, stream>>>(h2, as2, ad2, asr2, adt2, N, 1, 64);
    fill_k<<<cdiv((size_t)N, 256), 256, 0, stream>>>(amax2, -INFINITY, (size_t)N);
    fill_k<<<cdiv((size_t)N, 256), 256, 0, stream>>>(den2, 0.0f, (size_t)N);
    out_bias_k<<<cdiv((size_t)N * 64, 256), 256, 0, stream>>>(out, bias2, (size_t)N * 64, 64);
    size_t eh = (size_t)(E + N);
    edge_max_k<<<cdiv(eh, 256), 256, 0, stream>>>(src, dst, E, N, 1, asr2, adt2, amax2);
    edge_expsum_k<<<cdiv(eh, 256), 256, 0, stream>>>(src, dst, E, N, 1, asr2, adt2, amax2, ebuf, den2);
    size_t ec = (size_t)(E + N) * 64;
    edge_msg_k<<<cdiv(ec, 256), 256, 0, stream>>>(src, dst, E, N, 1, 64, ebuf, den2, h2, out);
  }
}